// IIRSVF_12575664243047
// MI455X (gfx1250) — compile-verified
//
#include <hip/hip_runtime.h>
#include <stdint.h>

// ---------------------------------------------------------------------------
// Time-varying SVF (2-state IIR) as a 3-phase chunked affine scan.
//   s_{t+1} = A_t s_t + b_t   (A_t = 2H_t - I, b_t = 2 gHB_t * x_t)
// Phase1: per-chunk affine composition (M,v)      [TDM-staged streaming]
// Phase2: per-batch scan over chunk affines -> chunk start states
// Phase3: replay chunks with known start state    [TDM-staged streaming + TDM store]
// Bandwidth-bound: ~340MB total traffic -> ~15us floor @ 23.3 TB/s.
// ---------------------------------------------------------------------------

typedef uint32_t u32;
typedef uint64_t u64;
typedef u32 u32x4 __attribute__((ext_vector_type(4)));
typedef u32 u32x8 __attribute__((ext_vector_type(8)));

#define T_LEN      131072
#define B_N        64
#define L_CHUNK    128                       // time steps per chunk (per thread)
#define CPB        (T_LEN / L_CHUNK)         // 1024 chunks per batch
#define CHUNKS_TOT (B_N * CPB)               // 65536
#define NTHREADS   256                       // 8 waves (wave32)
#define N_BLOCKS   (CHUNKS_TOT / NTHREADS)   // 256 blocks
#define TI         16                        // time steps per staged tile
#define NT         (L_CHUNK / TI)            // 8 tiles per chunk
#define ROWP       17                        // padded LDS row pitch (dwords) for width-16 tiles
#define MROWP      51                        // padded LDS row pitch for width-48 (mix) tiles

__device__ __forceinline__ u32 lds_off_u32(const void* p) {
  // generic pointer to LDS: low 32 bits are the LDS byte offset
  return (u32)(uintptr_t)p;
}

// Build a TDM descriptor (2D tile: rows x tile_w elements of 4B, row stride in
// elements) and issue tensor_load_to_lds / tensor_store_from_lds.
// pad=true inserts 1 dword of LDS padding every 16 dwords (loads only).
__device__ __forceinline__ void tdm_make(u32x4& g0, u32x8& g1, u64 gaddr_bytes,
                                         u32 ldsoff, u32 tile_w, u32 rows,
                                         u64 row_stride_elems, bool pad) {
  g0.x = 1u;                                   // count=1, user descriptor
  g0.y = ldsoff;                               // lds_addr [63:32]
  g0.z = (u32)gaddr_bytes;                     // global_addr low
  g0.w = (u32)(gaddr_bytes >> 32) | (2u << 30);// global_addr high | type=2 (image)

  g1.s0 = (2u << 16) | (pad ? ((1u << 20) | (3u << 22)) : 0u); // data_size=4B, pad 1dw/16dw
  g1.s1 = (tile_w & 0xFFFFu) << 16;            // tensor_dim0[15:0]
  g1.s2 = (tile_w >> 16) | ((rows & 0xFFFFu) << 16); // tensor_dim0 hi | tensor_dim1 lo
  g1.s3 = (rows >> 16) | (tile_w << 16);       // tensor_dim1 hi | tile_dim0
  g1.s4 = rows & 0xFFFFu;                      // tile_dim1 (tile_dim2=0)
  g1.s5 = (u32)row_stride_elems;               // tensor_dim0_stride low 32
  g1.s6 = (u32)(row_stride_elems >> 32) & 0xFFFFu; // stride[47:32], dim1_stride=0
  g1.s7 = 0u;
}

__device__ __forceinline__ void tdm_load(u64 ga, u32 ldso, u32 w, u32 rows,
                                         u64 stride, bool pad) {
  u32x4 g0; u32x8 g1;
  tdm_make(g0, g1, ga, ldso, w, rows, stride, pad);
  asm volatile("tensor_load_to_lds %0, %1" :: "s"(g0), "s"(g1) : "memory");
}

__device__ __forceinline__ void tdm_store(u64 ga, u32 ldso, u32 w, u32 rows,
                                          u64 stride) {
  u32x4 g0; u32x8 g1;
  tdm_make(g0, g1, ga, ldso, w, rows, stride, false);
  asm volatile("tensor_store_from_lds %0, %1" :: "s"(g0), "s"(g1) : "memory");
}

// -------------------------- Phase 1 ----------------------------------------
__global__ __launch_bounds__(NTHREADS)
void svf_phase1(const float* __restrict__ audio, const float* __restrict__ cutoff,
                const float* __restrict__ reso, const int* __restrict__ srp,
                float* __restrict__ aff) {
  __shared__ float lA[2][ROWP * NTHREADS];
  __shared__ float lC[2][ROWP * NTHREADS];
  __shared__ float lR[2][ROWP * NTHREADS];

  const int tid = threadIdx.x;
  const int bid = blockIdx.x;
  const float sr   = (float)srp[0];
  const float kk   = 3.14159265358979f / sr;
  const float cmax = sr * 0.5f - 1e-4f;
  const u64 span0 = (u64)bid * (NTHREADS * L_CHUNK);  // first time element of block

  if (tid < 32) {   // wave0 drives the TDM (EXEC ignored by TDM)
    tdm_load((u64)(uintptr_t)audio  + span0 * 4, lds_off_u32(&lA[0][0]), TI, NTHREADS, L_CHUNK, true);
    tdm_load((u64)(uintptr_t)cutoff + span0 * 4, lds_off_u32(&lC[0][0]), TI, NTHREADS, L_CHUNK, true);
    tdm_load((u64)(uintptr_t)reso   + span0 * 4, lds_off_u32(&lR[0][0]), TI, NTHREADS, L_CHUNK, true);
  }

  float M00 = 1.f, M01 = 0.f, M10 = 0.f, M11 = 1.f, V0 = 0.f, V1 = 0.f;

  for (int ti = 0; ti < NT; ++ti) {
    if (tid < 32) {
      if (ti + 1 < NT) {  // double-buffer: prefetch next tile
        const u64 e = span0 + (u64)(ti + 1) * TI;
        const int nb = (ti + 1) & 1;
        tdm_load((u64)(uintptr_t)audio  + e * 4, lds_off_u32(&lA[nb][0]), TI, NTHREADS, L_CHUNK, true);
        tdm_load((u64)(uintptr_t)cutoff + e * 4, lds_off_u32(&lC[nb][0]), TI, NTHREADS, L_CHUNK, true);
        tdm_load((u64)(uintptr_t)reso   + e * 4, lds_off_u32(&lR[nb][0]), TI, NTHREADS, L_CHUNK, true);
        __builtin_amdgcn_s_wait_tensorcnt(3);   // current tile complete
      } else {
        __builtin_amdgcn_s_wait_tensorcnt(0);
      }
    }
    __syncthreads();
    const int cb = ti & 1;
#pragma unroll
    for (int i = 0; i < TI; ++i) {
      float x = lA[cb][tid * ROWP + i];
      float c = lC[cb][tid * ROWP + i];
      float r = lR[cb][tid * ROWP + i];
      c = fminf(fmaxf(c, 1e-4f), cmax);
      r = fmaxf(r, 1e-4f);
      const float g    = __tanf(kk * c);
      const float twoR = 1.0f / r;
      const float Tc   = 1.0f / fmaf(g, g + twoR, 1.0f);
      const float t2   = 2.0f * Tc;
      const float A00 = t2 - 1.0f;
      const float A10 = t2 * g;
      const float A01 = -A10;
      const float A11 = t2 * fmaf(twoR, g, 1.0f) - 1.0f;
      const float b0  = A10 * x;   // 2 g Tc x
      const float b1  = g * b0;    // 2 g^2 Tc x
      // (M,V) <- A ∘ (M,V)
      const float n00 = A00 * M00 + A01 * M10;
      const float n01 = A00 * M01 + A01 * M11;
      const float n10 = A10 * M00 + A11 * M10;
      const float n11 = A10 * M01 + A11 * M11;
      const float nv0 = A00 * V0 + A01 * V1 + b0;
      const float nv1 = A10 * V0 + A11 * V1 + b1;
      M00 = n00; M01 = n01; M10 = n10; M11 = n11; V0 = nv0; V1 = nv1;
    }
    __syncthreads();
  }

  const int cid = bid * NTHREADS + tid;
  aff[0 * CHUNKS_TOT + cid] = M00;
  aff[1 * CHUNKS_TOT + cid] = M01;
  aff[2 * CHUNKS_TOT + cid] = M10;
  aff[3 * CHUNKS_TOT + cid] = M11;
  aff[4 * CHUNKS_TOT + cid] = V0;
  aff[5 * CHUNKS_TOT + cid] = V1;
}

// -------------------------- Phase 2 ----------------------------------------
// One block per batch: scan 1024 chunk affines (4 per thread + Hillis-Steele
// over 256 thread-local compositions), emit start state per chunk.
__global__ __launch_bounds__(NTHREADS)
void svf_phase2(const float* __restrict__ aff, float* __restrict__ sstart) {
  __shared__ float s00[NTHREADS], s01[NTHREADS], s10[NTHREADS], s11[NTHREADS];
  __shared__ float sv0[NTHREADS], sv1[NTHREADS];
  const int b = blockIdx.x;
  const int tid = threadIdx.x;
  const int base = b * CPB + tid * 4;

  float a00[4], a01[4], a10[4], a11[4], av0[4], av1[4];
#pragma unroll
  for (int j = 0; j < 4; ++j) {
    const int cid = base + j;
    a00[j] = aff[0 * CHUNKS_TOT + cid];
    a01[j] = aff[1 * CHUNKS_TOT + cid];
    a10[j] = aff[2 * CHUNKS_TOT + cid];
    a11[j] = aff[3 * CHUNKS_TOT + cid];
    av0[j] = aff[4 * CHUNKS_TOT + cid];
    av1[j] = aff[5 * CHUNKS_TOT + cid];
  }
  // local composition of 4 chunks: cur = a3 ∘ a2 ∘ a1 ∘ a0
  float M00 = a00[0], M01 = a01[0], M10 = a10[0], M11 = a11[0], V0 = av0[0], V1 = av1[0];
#pragma unroll
  for (int j = 1; j < 4; ++j) {
    const float n00 = a00[j] * M00 + a01[j] * M10;
    const float n01 = a00[j] * M01 + a01[j] * M11;
    const float n10 = a10[j] * M00 + a11[j] * M10;
    const float n11 = a10[j] * M01 + a11[j] * M11;
    const float nv0 = a00[j] * V0 + a01[j] * V1 + av0[j];
    const float nv1 = a10[j] * V0 + a11[j] * V1 + av1[j];
    M00 = n00; M01 = n01; M10 = n10; M11 = n11; V0 = nv0; V1 = nv1;
  }
  s00[tid] = M00; s01[tid] = M01; s10[tid] = M10; s11[tid] = M11;
  sv0[tid] = V0;  sv1[tid] = V1;

  for (int d = 1; d < NTHREADS; d <<= 1) {
    __syncthreads();
    float p00 = 0, p01 = 0, p10 = 0, p11 = 0, pv0 = 0, pv1 = 0;
    const bool has = tid >= d;
    if (has) {
      p00 = s00[tid - d]; p01 = s01[tid - d]; p10 = s10[tid - d]; p11 = s11[tid - d];
      pv0 = sv0[tid - d]; pv1 = sv1[tid - d];
    }
    __syncthreads();
    if (has) {            // cur = cur ∘ prev  (prev is earlier in time)
      const float n00 = M00 * p00 + M01 * p10;
      const float n01 = M00 * p01 + M01 * p11;
      const float n10 = M10 * p00 + M11 * p10;
      const float n11 = M10 * p01 + M11 * p11;
      const float nv0 = M00 * pv0 + M01 * pv1 + V0;
      const float nv1 = M10 * pv0 + M11 * pv1 + V1;
      M00 = n00; M01 = n01; M10 = n10; M11 = n11; V0 = nv0; V1 = nv1;
      s00[tid] = M00; s01[tid] = M01; s10[tid] = M10; s11[tid] = M11;
      sv0[tid] = V0;  sv1[tid] = V1;
    }
  }
  __syncthreads();
  // exclusive prefix applied to zero initial state -> just v of inclusive[tid-1]
  float S0 = 0.f, S1 = 0.f;
  if (tid > 0) { S0 = sv0[tid - 1]; S1 = sv1[tid - 1]; }
#pragma unroll
  for (int j = 0; j < 4; ++j) {
    const int cid = base + j;
    sstart[2 * cid + 0] = S0;
    sstart[2 * cid + 1] = S1;
    const float n0 = a00[j] * S0 + a01[j] * S1 + av0[j];
    const float n1 = a10[j] * S0 + a11[j] * S1 + av1[j];
    S0 = n0; S1 = n1;
  }
}

// -------------------------- Phase 3 ----------------------------------------
__global__ __launch_bounds__(NTHREADS)
void svf_phase3(const float* __restrict__ audio, const float* __restrict__ cutoff,
                const float* __restrict__ reso, const float* __restrict__ mixp,
                const float* __restrict__ sstart, const int* __restrict__ srp,
                float* __restrict__ out) {
  __shared__ float lA[2][ROWP * NTHREADS];
  __shared__ float lC[2][ROWP * NTHREADS];
  __shared__ float lR[2][ROWP * NTHREADS];
  __shared__ float lM[2][MROWP * NTHREADS];
  __shared__ float lO[TI * NTHREADS];          // unpadded: TDM store reads LDS contiguously

  const int tid = threadIdx.x;
  const int bid = blockIdx.x;
  const float sr   = (float)srp[0];
  const float kk   = 3.14159265358979f / sr;
  const float cmax = sr * 0.5f - 1e-4f;
  const u64 span0 = (u64)bid * (NTHREADS * L_CHUNK);
  const int cid = bid * NTHREADS + tid;
  float s0 = sstart[2 * cid + 0];
  float s1 = sstart[2 * cid + 1];

  if (tid < 32) {
    tdm_load((u64)(uintptr_t)audio  + span0 * 4,  lds_off_u32(&lA[0][0]), TI,     NTHREADS, L_CHUNK,     true);
    tdm_load((u64)(uintptr_t)cutoff + span0 * 4,  lds_off_u32(&lC[0][0]), TI,     NTHREADS, L_CHUNK,     true);
    tdm_load((u64)(uintptr_t)reso   + span0 * 4,  lds_off_u32(&lR[0][0]), TI,     NTHREADS, L_CHUNK,     true);
    tdm_load((u64)(uintptr_t)mixp   + span0 * 12, lds_off_u32(&lM[0][0]), TI * 3, NTHREADS, L_CHUNK * 3, true);
  }

  for (int ti = 0; ti < NT; ++ti) {
    if (tid < 32) {
      if (ti + 1 < NT) {
        const u64 e = span0 + (u64)(ti + 1) * TI;
        const int nb = (ti + 1) & 1;
        tdm_load((u64)(uintptr_t)audio  + e * 4,  lds_off_u32(&lA[nb][0]), TI,     NTHREADS, L_CHUNK,     true);
        tdm_load((u64)(uintptr_t)cutoff + e * 4,  lds_off_u32(&lC[nb][0]), TI,     NTHREADS, L_CHUNK,     true);
        tdm_load((u64)(uintptr_t)reso   + e * 4,  lds_off_u32(&lR[nb][0]), TI,     NTHREADS, L_CHUNK,     true);
        tdm_load((u64)(uintptr_t)mixp   + e * 12, lds_off_u32(&lM[nb][0]), TI * 3, NTHREADS, L_CHUNK * 3, true);
        __builtin_amdgcn_s_wait_tensorcnt(4);  // prior store + current-tile loads done
      } else {
        __builtin_amdgcn_s_wait_tensorcnt(0);
      }
    }
    __syncthreads();
    const int cb = ti & 1;
#pragma unroll
    for (int i = 0; i < TI; ++i) {
      float x = lA[cb][tid * ROWP + i];
      float c = lC[cb][tid * ROWP + i];
      float r = lR[cb][tid * ROWP + i];
      const int p = i * 3;   // padded mix row: +1 dword every 16 dwords
      const float m0 = lM[cb][tid * MROWP + (p + 0) + ((p + 0) >> 4)];
      const float m1 = lM[cb][tid * MROWP + (p + 1) + ((p + 1) >> 4)];
      const float m2 = lM[cb][tid * MROWP + (p + 2) + ((p + 2) >> 4)];
      c = fminf(fmaxf(c, 1e-4f), cmax);
      r = fmaxf(r, 1e-4f);
      const float g    = __tanf(kk * c);
      const float twoR = 1.0f / r;
      const float Tc   = 1.0f / fmaf(g, g + twoR, 1.0f);
      const float gTc  = g * Tc;
      // Y = gHB*x + H*s ; H = Tc*[[1,-g],[g,twoR*g+1]]
      const float Y0 = gTc * x + Tc * s0 - gTc * s1;
      const float Y1 = g * (gTc * x) + gTc * s0 + Tc * fmaf(twoR, g, 1.0f) * s1;
      s0 = 2.0f * Y0 - s0;
      s1 = 2.0f * Y1 - s1;
      const float yhp = x - twoR * Y0 - Y1;
      lO[tid * TI + i] = twoR * m0 * Y0 + m1 * Y1 + m2 * yhp;
    }
    __syncthreads();
    if (tid < 32) {
      const u64 e = span0 + (u64)ti * TI;
      tdm_store((u64)(uintptr_t)out + e * 4, lds_off_u32(&lO[0]), TI, NTHREADS, L_CHUNK);
    }
  }
  // implicit S_WAIT_IDLE at S_ENDPGM drains the final TDM store
}

// ---------------------------------------------------------------------------
extern "C" void kernel_launch(void* const* d_in, const int* in_sizes, int n_in,
                              void* d_out, int out_size, void* d_ws, size_t ws_size,
                              hipStream_t stream) {
  (void)in_sizes; (void)n_in; (void)out_size; (void)ws_size;
  const float* audio  = (const float*)d_in[0];
  const float* cutoff = (const float*)d_in[1];
  const float* reso   = (const float*)d_in[2];
  const float* mixp   = (const float*)d_in[3];
  const int*   srp    = (const int*)d_in[4];
  float* out = (float*)d_out;

  float* aff    = (float*)d_ws;                 // 6 * 65536 floats
  float* sstart = aff + 6 * CHUNKS_TOT;         // 2 * 65536 floats

  svf_phase1<<<N_BLOCKS, NTHREADS, 0, stream>>>(audio, cutoff, reso, srp, aff);
  svf_phase2<<<B_N, NTHREADS, 0, stream>>>(aff, sstart);
  svf_phase3<<<N_BLOCKS, NTHREADS, 0, stream>>>(audio, cutoff, reso, mixp, sstart, srp, out);
}